// VoxelWithPointProjectionV2_57741540327457
// MI455X (gfx1250) — compile-verified
//
#include <hip/hip_runtime.h>
#include <hip/hip_bf16.h>

typedef __attribute__((ext_vector_type(2))) float v2f;
typedef __attribute__((ext_vector_type(8))) float v8f;

#define IMH 192
#define IMW 640
#define HWSZ (IMH * IMW)
#define CDIM 64
#define BN_EPS 1e-3f

// ---------------------------------------------------------------------------
// f32 WMMA 16x16x4 wrapper.  D = A(16x4) x B(4x16) + C(16x16), all fp32.
// A layout: lane<16 -> row M=lane, v0=K0,v1=K1 ; lane>=16 -> row M=lane-16, v0=K2,v1=K3
// B layout: lane<16 -> col N=lane, v0=K0,v1=K1 ; lane>=16 -> col N=lane-16, v0=K2,v1=K3
// C/D layout: col N=lane&15; VGPR i = row (lane<16 ? i : 8+i)
// ---------------------------------------------------------------------------
__device__ __forceinline__ v8f wmma4(v2f a, v2f b, v8f c) {
  return __builtin_amdgcn_wmma_f32_16x16x4_f32(false, a, false, b, (short)0, c,
                                               false, false);
}

// Generic 16xK row-major A  times  KxN row-major B, producing NT 16x16 tiles.
template <int NT>
__device__ __forceinline__ void gemm16(const float* A, int lda,
                                       const float* __restrict__ B, int ldb,
                                       int K, v8f* acc, int lane) {
  const int m  = lane & 15;
  const int kh = (lane >> 4) << 1;  // 0 or 2
  for (int k0 = 0; k0 < K; k0 += 4) {
    v2f a;
    a.x = A[m * lda + k0 + kh];
    a.y = A[m * lda + k0 + kh + 1];
#pragma unroll
    for (int t = 0; t < NT; ++t) {
      v2f b;
      b.x = B[(k0 + kh) * ldb + t * 16 + m];
      b.y = B[(k0 + kh + 1) * ldb + t * 16 + m];
      acc[t] = wmma4(a, b, acc[t]);
    }
  }
}

// ---------------------------------------------------------------------------
// Kernel 1: value = image_feat^T @ Wv + bv     (HW x 64)
// image_feat is (64, HW);  one wave per 16-hw-row tile.
// feat is read exactly once -> non-temporal loads keep L2 free for `value`,
// which must stay L2-resident (31.5 MB of 192 MB) for kernel 2's gathers.
// ---------------------------------------------------------------------------
__global__ __launch_bounds__(256) void value_proj_kernel(
    const float* __restrict__ feat, const float* __restrict__ Wv,
    const float* __restrict__ bv, float* __restrict__ value) {
  const int lane = threadIdx.x & 31;
  const int wave = threadIdx.x >> 5;
  const int tile = blockIdx.x * (blockDim.x >> 5) + wave;
  const int hw0  = tile * 16;
  if (hw0 >= HWSZ) return;

  const int m  = lane & 15;
  const int kh = (lane >> 4) << 1;
  v8f acc[4] = {v8f{}, v8f{}, v8f{}, v8f{}};

  for (int k0 = 0; k0 < CDIM; k0 += 4) {
    v2f a;  // A[m][k] = feat[k*HW + hw0 + m]  (coalesced in m, streamed NT)
    a.x = __builtin_nontemporal_load(&feat[(k0 + kh) * HWSZ + hw0 + m]);
    a.y = __builtin_nontemporal_load(&feat[(k0 + kh + 1) * HWSZ + hw0 + m]);
#pragma unroll
    for (int t = 0; t < 4; ++t) {
      v2f b;
      b.x = Wv[(k0 + kh) * CDIM + t * 16 + m];
      b.y = Wv[(k0 + kh + 1) * CDIM + t * 16 + m];
      acc[t] = wmma4(a, b, acc[t]);
    }
  }

  const int mh = (lane >> 4) << 3;
#pragma unroll
  for (int t = 0; t < 4; ++t)
#pragma unroll
    for (int i = 0; i < 8; ++i)
      value[(hw0 + mh + i) * CDIM + t * 16 + m] = acc[t][i] + bv[t * 16 + m];
}

// ---------------------------------------------------------------------------
// Kernel 2: fully fused per-point pipeline, one wave32 per 16-point tile.
// ---------------------------------------------------------------------------
#define TS 68    // stride of staging tiles (16x64 + pad, bank-conflict free)
#define FS 132   // stride of fuse tile     (16x128 + pad)

__global__ __launch_bounds__(32) void fused_point_kernel(
    const float* __restrict__ pf, const float* __restrict__ value,
    const int* __restrict__ igrid, const unsigned char* __restrict__ pmask,
    const float* __restrict__ Wk, const float* __restrict__ bk,
    const float* __restrict__ Wt, const float* __restrict__ bt,
    const float* __restrict__ Wo, const float* __restrict__ bo,
    const float* __restrict__ Wa, const float* __restrict__ ba,
    const float* __restrict__ Wout, const float* __restrict__ bout,
    const float* __restrict__ Wf, const float* __restrict__ bf,
    const float* __restrict__ bnk_g, const float* __restrict__ bnk_b,
    const float* __restrict__ bnk_m, const float* __restrict__ bnk_v,
    const float* __restrict__ bnt_g, const float* __restrict__ bnt_b,
    const float* __restrict__ bnt_m, const float* __restrict__ bnt_v,
    const float* __restrict__ bnf_g, const float* __restrict__ bnf_b,
    const float* __restrict__ bnf_m, const float* __restrict__ bnf_v,
    float* __restrict__ out, int N) {
  __shared__ float P[16 * TS];  // staged pf tile (read by Wk and Wt GEMMs)
  __shared__ float T[16 * TS];  // q staging, later attn_in staging
  __shared__ float S[16 * FS];  // off/logits staging, later fuse (16x128)

  const int lane = threadIdx.x & 31;
  const int r0   = blockIdx.x * 16;
  const int m    = lane & 15;
  const int mh   = (lane >> 4) << 3;

  // ---- 0. stage the 16x64 pf tile into LDS with coalesced b128 loads ------
  {
    const float4* pf4 = reinterpret_cast<const float4*>(pf);
#pragma unroll
    for (int it = 0; it < 8; ++it) {
      const int idx  = it * 32 + lane;       // 256 float4s total
      const int row  = idx >> 4;             // 0..15
      const int col4 = idx & 15;             // 0..15
      int gr = r0 + row;
      if (gr >= N) gr = N - 1;               // tail clamp (in-bounds read)
      const float4 v = pf4[(size_t)gr * 16 + col4];
      float* d = &P[row * TS + col4 * 4];
      d[0] = v.x; d[1] = v.y; d[2] = v.z; d[3] = v.w;
    }
  }
  __syncthreads();  // single-wave WG: hardware NOP, compiler memory fence

  // ---- 1. q = BN_k(pf @ Wk + bk) -> T[16x64] -------------------------------
  {
    v8f acc[4] = {v8f{}, v8f{}, v8f{}, v8f{}};
    gemm16<4>(P, TS, Wk, CDIM, CDIM, acc, lane);
#pragma unroll
    for (int t = 0; t < 4; ++t) {
      const int c = t * 16 + m;
      const float s  = bnk_g[c] * rsqrtf(bnk_v[c] + BN_EPS);
      const float sh = bnk_b[c] - bnk_m[c] * s;
#pragma unroll
      for (int i = 0; i < 8; ++i)
        T[(mh + i) * TS + c] = (acc[t][i] + bk[c]) * s + sh;
    }
  }
  __syncthreads();

  // ---- 2. off = q@Wo + bo (16x32), logits = q@Wa + ba (16x16) -> S ---------
  {
    v8f acco[2] = {v8f{}, v8f{}};
    gemm16<2>(T, TS, Wo, 32, CDIM, acco, lane);
    v8f acca[1] = {v8f{}};
    gemm16<1>(T, TS, Wa, 16, CDIM, acca, lane);
#pragma unroll
    for (int t = 0; t < 2; ++t)
#pragma unroll
      for (int i = 0; i < 8; ++i)
        S[(mh + i) * FS + t * 16 + m] = acco[t][i] + bo[t * 16 + m];
#pragma unroll
    for (int i = 0; i < 8; ++i)
      S[(mh + i) * FS + 32 + m] = acca[0][i] + ba[m];
  }
  __syncthreads();

  // ---- 3. deformable sampling: 2 lanes per point, 2 heads per lane --------
  {
    const int p = lane >> 1;
    int r = r0 + p;
    if (r >= N) r = N - 1;
    const int gx = igrid[2 * r + 0];
    const int gy = igrid[2 * r + 1];
    const float* Srow = &S[p * FS];

#pragma unroll
    for (int hh = 0; hh < 2; ++hh) {
      const int h = ((lane & 1) << 1) + hh;
      // softmax over the 4 sampling points of this head
      float lg[4], w[4];
#pragma unroll
      for (int q = 0; q < 4; ++q) lg[q] = Srow[32 + h * 4 + q];
      float mx = fmaxf(fmaxf(lg[0], lg[1]), fmaxf(lg[2], lg[3]));
      float sum = 0.f;
#pragma unroll
      for (int q = 0; q < 4; ++q) { w[q] = __expf(lg[q] - mx); sum += w[q]; }
      const float inv = 1.f / sum;

      float a4[16];
#pragma unroll
      for (int c = 0; c < 16; ++c) a4[c] = 0.f;

#pragma unroll
      for (int q = 0; q < 4; ++q) {
        const float aw = w[q] * inv;
        const float px = (float)gx + Srow[h * 8 + q * 2 + 0] - 0.5f;
        const float py = (float)gy + Srow[h * 8 + q * 2 + 1] - 0.5f;
        const float fx = floorf(px), fy = floorf(py);
        const int x0 = (int)fx, y0 = (int)fy;
        const float wx = px - fx, wy = py - fy;
#pragma unroll
        for (int dy = 0; dy < 2; ++dy)
#pragma unroll
          for (int dx = 0; dx < 2; ++dx) {
            const int xi = x0 + dx, yi = y0 + dy;
            if (xi >= 0 && xi < IMW && yi >= 0 && yi < IMH) {
              const float cw = aw * (dx ? wx : 1.f - wx) * (dy ? wy : 1.f - wy);
              const float4* vp4 = reinterpret_cast<const float4*>(
                  value + ((size_t)yi * IMW + xi) * CDIM + h * 16);
#pragma unroll
              for (int qq = 0; qq < 4; ++qq) {
                const float4 v = vp4[qq];
                a4[qq * 4 + 0] = fmaf(cw, v.x, a4[qq * 4 + 0]);
                a4[qq * 4 + 1] = fmaf(cw, v.y, a4[qq * 4 + 1]);
                a4[qq * 4 + 2] = fmaf(cw, v.z, a4[qq * 4 + 2]);
                a4[qq * 4 + 3] = fmaf(cw, v.w, a4[qq * 4 + 3]);
              }
            }
          }
      }
#pragma unroll
      for (int c = 0; c < 16; ++c) T[p * TS + h * 16 + c] = a4[c];
    }
  }
  __syncthreads();

  // ---- 4. attn = (attn_in @ Wout + bout), mask, relu -> S cols 64..127 ----
  {
    v8f acc[4] = {v8f{}, v8f{}, v8f{}, v8f{}};
    gemm16<4>(T, TS, Wout, CDIM, CDIM, acc, lane);
#pragma unroll
    for (int t = 0; t < 4; ++t) {
      const int c = t * 16 + m;
#pragma unroll
      for (int i = 0; i < 8; ++i) {
        const int row = mh + i;
        int gr = r0 + row;
        if (gr >= N) gr = N - 1;
        float v = acc[t][i] + bout[c];
        if (!pmask[gr]) v = 0.f;
        S[row * FS + 64 + c] = fmaxf(v, 0.f);
      }
    }
  }

  // ---- 5. pts = BN_t(pf @ Wt + bt), relu -> S cols 0..63 ------------------
  {
    v8f acc[4] = {v8f{}, v8f{}, v8f{}, v8f{}};
    gemm16<4>(P, TS, Wt, CDIM, CDIM, acc, lane);
#pragma unroll
    for (int t = 0; t < 4; ++t) {
      const int c = t * 16 + m;
      const float s  = bnt_g[c] * rsqrtf(bnt_v[c] + BN_EPS);
      const float sh = bnt_b[c] - bnt_m[c] * s;
#pragma unroll
      for (int i = 0; i < 8; ++i)
        S[(mh + i) * FS + c] = fmaxf((acc[t][i] + bt[c]) * s + sh, 0.f);
    }
  }
  __syncthreads();

  // ---- 6. out = relu(BN_f(fuse @ Wf + bf)) --------------------------------
  {
    v8f acc[4] = {v8f{}, v8f{}, v8f{}, v8f{}};
    gemm16<4>(S, FS, Wf, CDIM, 128, acc, lane);
#pragma unroll
    for (int t = 0; t < 4; ++t) {
      const int c = t * 16 + m;
      const float s  = bnf_g[c] * rsqrtf(bnf_v[c] + BN_EPS);
      const float sh = bnf_b[c] - bnf_m[c] * s;
#pragma unroll
      for (int i = 0; i < 8; ++i) {
        const int row = r0 + mh + i;
        if (row < N)
          out[(size_t)row * CDIM + c] = fmaxf((acc[t][i] + bf[c]) * s + sh, 0.f);
      }
    }
  }
}

// ---------------------------------------------------------------------------
extern "C" void kernel_launch(void* const* d_in, const int* in_sizes, int n_in,
                              void* d_out, int out_size, void* d_ws,
                              size_t ws_size, hipStream_t stream) {
  const float* pf    = (const float*)d_in[0];
  const float* feat  = (const float*)d_in[1];
  const int* igrid   = (const int*)d_in[2];
  const unsigned char* pmask = (const unsigned char*)d_in[3];
  const float* Wk    = (const float*)d_in[4];
  const float* bk    = (const float*)d_in[5];
  const float* Wt    = (const float*)d_in[6];
  const float* bt    = (const float*)d_in[7];
  const float* Wo    = (const float*)d_in[8];
  const float* bo    = (const float*)d_in[9];
  const float* Wa    = (const float*)d_in[10];
  const float* ba    = (const float*)d_in[11];
  const float* Wv    = (const float*)d_in[12];
  const float* bv    = (const float*)d_in[13];
  const float* Wout  = (const float*)d_in[14];
  const float* bout  = (const float*)d_in[15];
  const float* Wf    = (const float*)d_in[16];
  const float* bf    = (const float*)d_in[17];
  const float* bnk_g = (const float*)d_in[18];
  const float* bnk_b = (const float*)d_in[19];
  const float* bnk_m = (const float*)d_in[20];
  const float* bnk_v = (const float*)d_in[21];
  const float* bnt_g = (const float*)d_in[22];
  const float* bnt_b = (const float*)d_in[23];
  const float* bnt_m = (const float*)d_in[24];
  const float* bnt_v = (const float*)d_in[25];
  const float* bnf_g = (const float*)d_in[26];
  const float* bnf_b = (const float*)d_in[27];
  const float* bnf_m = (const float*)d_in[28];
  const float* bnf_v = (const float*)d_in[29];

  const int N = in_sizes[0] / CDIM;
  float* value = (float*)d_ws;  // HW x 64 fp32 = 31.5 MB, L2-resident

  // Kernel 1: project image features to value tensor (WMMA GEMM).
  const int vtiles = HWSZ / 16;                  // 7680
  value_proj_kernel<<<(vtiles + 7) / 8, 256, 0, stream>>>(feat, Wv, bv, value);

  // Kernel 2: fused BN/GEMM/deform-attn/fuse pipeline, 1 wave per 16 points.
  const int ptiles = (N + 15) / 16;
  fused_point_kernel<<<ptiles, 32, 0, stream>>>(
      pf, value, igrid, pmask, Wk, bk, Wt, bt, Wo, bo, Wa, ba, Wout, bout, Wf,
      bf, bnk_g, bnk_b, bnk_m, bnk_v, bnt_g, bnt_b, bnt_m, bnt_v, bnf_g, bnf_b,
      bnf_m, bnf_v, (float*)d_out, N);
}